// S9Layer_28535762715091
// MI455X (gfx1250) — compile-verified
//
#include <hip/hip_runtime.h>
#include <hip/hip_bf16.h>

// ---------------------------------------------------------------------------
// S4ND-style 2D SSM layer on gfx1250, recast as three complex GEMMs:
//   y1 = T(k0) @ U            (causal conv along L1; Toeplitz matmul)
//   y2 = y1 @ T(k1)^T         (causal conv along L2)
//   out = W @ y2 (per pixel)  (complex channel mixing)
// GEMMs run on V_WMMA_F32_16X16X4_F32 (wave32 matrix pipe, fp32 fidelity).
// Toeplitz operands are synthesized from a ZERO-PADDED LDS image of the 1-D
// kernel (sk[0..127]=0, sk[128..255]=k), so every fragment gather is an
// unconditional ds_load -- no EXEC-masked conditional loads in the hot loop.
// Each pass computes TWO 16x16 N-tiles sharing one A-fragment (8 WMMA per
// K-step against 4 LDS + 8 global loads).
// ---------------------------------------------------------------------------

#define D_MODEL 128
#define N_STATE 64
#define LEN     128          // L1 == L2 == 128
#define BATCH   4
#define IMG     (LEN * LEN)  // 16384 pixels per (b,c) image
#define KPAD    (2 * LEN)    // zero-padded LDS kernel image

typedef float v2f __attribute__((ext_vector_type(2)));
typedef float v8f __attribute__((ext_vector_type(8)));

// D = A*B + C  (16x16 f32 out, A 16x4, B 4x16)
__device__ __forceinline__ v8f wmma_p(v2f a, v2f b, v8f c) {
  return __builtin_amdgcn_wmma_f32_16x16x4_f32(false, a, false, b, (short)0, c,
                                               false, false);
}

// ---------------------------------------------------------------------------
// Stage 1: SSM kernels k[dir][c][l] = sum_n (C*B)[n] * exp((A*dt)[n] * l)
// ---------------------------------------------------------------------------
__global__ __launch_bounds__(LEN) void ssm_k_kernel(
    const float* __restrict__ logA,   // (2, D, N)
    const float* __restrict__ Aimag,  // (2, N)
    const float* __restrict__ logdt,  // (2, D)
    const float* __restrict__ Bre, const float* __restrict__ Bim,  // (2, D, N)
    const float* __restrict__ Cre, const float* __restrict__ Cim,  // (2, D, N)
    float* __restrict__ k_re, float* __restrict__ k_im)            // (2, D, L)
{
  const int dir = blockIdx.x >> 7;
  const int ch  = blockIdx.x & 127;
  const int l   = threadIdx.x;
  const float fl = (float)l;
  const float dt = __expf(logdt[dir * D_MODEL + ch]);
  const int base = (dir * D_MODEL + ch) * N_STATE;

  float sre = 0.f, sim = 0.f;
  for (int n = 0; n < N_STATE; ++n) {
    const float ar = -__expf(logA[base + n]);
    const float ai = Aimag[dir * N_STATE + n];
    const float br = Bre[base + n], bi = Bim[base + n];
    const float cr = Cre[base + n], ci = Cim[base + n];
    const float cbr = cr * br - ci * bi;   // (C*B) complex
    const float cbi = cr * bi + ci * br;
    const float mag = __expf(ar * dt * fl);
    float sn, cs;
    __sincosf(ai * dt * fl, &sn, &cs);
    const float pre = mag * cs, pim = mag * sn;   // exp(A*dt*l)
    sre = fmaf(cbr, pre, fmaf(-cbi, pim, sre));
    sim = fmaf(cbr, pim, fmaf(cbi, pre, sim));
  }
  k_re[(dir * D_MODEL + ch) * LEN + l] = sre;
  k_im[(dir * D_MODEL + ch) * LEN + l] = sim;
}

// Stage the 1-D kernel into zero-padded LDS: sk[0..127]=0, sk[128..255]=k[ch].
__device__ __forceinline__ void stage_k_padded(float* skre, float* skim,
                                               const float* __restrict__ kre,
                                               const float* __restrict__ kim,
                                               int ch, int tid) {
  const float vr = (tid >= LEN) ? kre[ch * LEN + (tid - LEN)] : 0.f;
  const float vi = (tid >= LEN) ? kim[ch * LEN + (tid - LEN)] : 0.f;
  skre[tid] = vr;
  skim[tid] = vi;
  __syncthreads();
}

// ---------------------------------------------------------------------------
// Stage 2: causal conv along rows:  y1[i,j] = sum_{m<=i} k0[i-m] * u[m,j]
// One 256-thread workgroup (8 waves) per (b,c) image; wave w owns M-tile w.
// ---------------------------------------------------------------------------
__global__ __launch_bounds__(256) void conv_rows_kernel(
    const float* __restrict__ ure, const float* __restrict__ uim,
    const float* __restrict__ kre, const float* __restrict__ kim,  // (D, L)
    float* __restrict__ yre, float* __restrict__ yim)
{
  __shared__ float skre[KPAD];
  __shared__ float skim[KPAD];
  const int img = blockIdx.x;     // b*128 + c
  const int tid = threadIdx.x;
  stage_k_padded(skre, skim, kre, kim, img & 127, tid);

  const int wave = tid >> 5, lane = tid & 31;
  const int hi = lane >> 4;      // lane-half selects K pair (A-frag layout)
  const int ln = lane & 15;
  const int i  = wave * 16 + ln; // output row (A-frag M index)
  const size_t ib = (size_t)img * IMG;
  const float* Ur = ure + ib;
  const float* Ui = uim + ib;
  float* Yr = yre + ib;
  float* Yi = yim + ib;

  for (int tn = 0; tn < 8; tn += 2) {
    const int n0 = tn * 16 + ln;         // two output col tiles share A frags
    const int n1 = n0 + 16;
    v8f ar0 = {}, ai0 = {}, ar1 = {}, ai1 = {};
    for (int kk = 0; kk < LEN; kk += 4) {
      const int m0 = kk + hi * 2;        // K indices held by this lane-half
      const int m1 = m0 + 1;
      v2f fAr, fAi, fAiN;
      fAr.x = skre[LEN + i - m0];        // padded: always in-range, no branch
      fAr.y = skre[LEN + i - m1];
      fAi.x = skim[LEN + i - m0];
      fAi.y = skim[LEN + i - m1];
      fAiN = -fAi;                       // VALU negate (no f32 neg_a modifier)
      v2f fBr0, fBi0, fBr1, fBi1;
      fBr0.x = Ur[m0 * LEN + n0];        // coalesced across lanes (n)
      fBr0.y = Ur[m1 * LEN + n0];
      fBi0.x = Ui[m0 * LEN + n0];
      fBi0.y = Ui[m1 * LEN + n0];
      fBr1.x = Ur[m0 * LEN + n1];
      fBr1.y = Ur[m1 * LEN + n1];
      fBi1.x = Ui[m0 * LEN + n1];
      fBi1.y = Ui[m1 * LEN + n1];
      __builtin_prefetch(&Ur[(m0 + 4) * LEN + n0], 0, 1);  // speculative
      __builtin_prefetch(&Ui[(m0 + 4) * LEN + n0], 0, 1);
      ar0 = wmma_p(fAr, fBr0, ar0);      // re += Ar*Br
      ar0 = wmma_p(fAiN, fBi0, ar0);     // re -= Ai*Bi
      ai0 = wmma_p(fAr, fBi0, ai0);      // im += Ar*Bi
      ai0 = wmma_p(fAi, fBr0, ai0);      // im += Ai*Br
      ar1 = wmma_p(fAr, fBr1, ar1);
      ar1 = wmma_p(fAiN, fBi1, ar1);
      ai1 = wmma_p(fAr, fBi1, ai1);
      ai1 = wmma_p(fAi, fBr1, ai1);
    }
    const int rbase = wave * 16 + hi * 8; // C/D layout: V r -> M=r / M=8+r
#pragma unroll
    for (int r = 0; r < 8; ++r) {
      const int row = rbase + r;
      Yr[row * LEN + n0] = ar0[r];
      Yi[row * LEN + n0] = ai0[r];
      Yr[row * LEN + n1] = ar1[r];
      Yi[row * LEN + n1] = ai1[r];
    }
  }
}

// ---------------------------------------------------------------------------
// Stage 3: causal conv along cols:  y2[i,j] = sum_{m<=j} y1[i,m] * k1[j-m]
// GEMM form: A = y1 (i x m, adjacent-pair loads), B Toeplitz from padded LDS.
// Two N-tiles share the (strided) global A-fragment loads.
// ---------------------------------------------------------------------------
__global__ __launch_bounds__(256) void conv_cols_kernel(
    const float* __restrict__ y1re, const float* __restrict__ y1im,
    const float* __restrict__ kre, const float* __restrict__ kim,  // (D, L)
    float* __restrict__ y2re, float* __restrict__ y2im)
{
  __shared__ float skre[KPAD];
  __shared__ float skim[KPAD];
  const int img = blockIdx.x;
  const int tid = threadIdx.x;
  stage_k_padded(skre, skim, kre, kim, img & 127, tid);

  const int wave = tid >> 5, lane = tid & 31;
  const int hi = lane >> 4;
  const int ln = lane & 15;
  const int i  = wave * 16 + ln;        // output row == y1 row
  const size_t ib = (size_t)img * IMG;
  const float* Ar = y1re + ib;
  const float* Ai = y1im + ib;
  float* Yr = y2re + ib;
  float* Yi = y2im + ib;

  for (int tn = 0; tn < 8; tn += 2) {
    const int j0 = tn * 16 + ln;
    const int j1 = j0 + 16;
    v8f ar0 = {}, ai0 = {}, ar1 = {}, ai1 = {};
    for (int kk = 0; kk < LEN; kk += 4) {
      const int m0 = kk + hi * 2;
      const int m1 = m0 + 1;
      v2f fAr, fAi, fAiN;
      fAr.x = Ar[i * LEN + m0];          // adjacent -> b64-mergeable
      fAr.y = Ar[i * LEN + m1];
      fAi.x = Ai[i * LEN + m0];
      fAi.y = Ai[i * LEN + m1];
      fAiN = -fAi;
      v2f fBr0, fBi0, fBr1, fBi1;
      fBr0.x = skre[LEN + j0 - m0];      // padded Toeplitz: unconditional
      fBr0.y = skre[LEN + j0 - m1];
      fBi0.x = skim[LEN + j0 - m0];
      fBi0.y = skim[LEN + j0 - m1];
      fBr1.x = skre[LEN + j1 - m0];
      fBr1.y = skre[LEN + j1 - m1];
      fBi1.x = skim[LEN + j1 - m0];
      fBi1.y = skim[LEN + j1 - m1];
      __builtin_prefetch(&Ar[i * LEN + m0 + 4], 0, 1);
      __builtin_prefetch(&Ai[i * LEN + m0 + 4], 0, 1);
      ar0 = wmma_p(fAr, fBr0, ar0);
      ar0 = wmma_p(fAiN, fBi0, ar0);
      ai0 = wmma_p(fAr, fBi0, ai0);
      ai0 = wmma_p(fAi, fBr0, ai0);
      ar1 = wmma_p(fAr, fBr1, ar1);
      ar1 = wmma_p(fAiN, fBi1, ar1);
      ai1 = wmma_p(fAr, fBi1, ai1);
      ai1 = wmma_p(fAi, fBr1, ai1);
    }
    const int rbase = wave * 16 + hi * 8;
#pragma unroll
    for (int r = 0; r < 8; ++r) {
      const int row = rbase + r;
      Yr[row * LEN + j0] = ar0[r];
      Yi[row * LEN + j0] = ai0[r];
      Yr[row * LEN + j1] = ar1[r];
      Yi[row * LEN + j1] = ai1[r];
    }
  }
}

// ---------------------------------------------------------------------------
// Stage 4: complex channel mixing: out[b,d,p] = sum_c W[d,c] * y2[b,c,p]
// Per batch: W(128x128) @ Y2(128x16384). Two pixel tiles share W fragments.
// Output written as interleaved complex64 (re, im).
// ---------------------------------------------------------------------------
__global__ __launch_bounds__(256) void mix_kernel(
    const float* __restrict__ y2re, const float* __restrict__ y2im,
    const float* __restrict__ Wre, const float* __restrict__ Wim,  // (D, D)
    float* __restrict__ out)                                       // complex
{
  const int b     = blockIdx.x >> 7;
  const int chunk = blockIdx.x & 127;   // 128 pixels per chunk
  const int tid = threadIdx.x;
  const int wave = tid >> 5, lane = tid & 31;
  const int hi = lane >> 4;
  const int ln = lane & 15;
  const int d  = wave * 16 + ln;        // output channel (A-frag M index)
  const int pixBase = chunk * 128;

  for (int tn = 0; tn < 8; tn += 2) {
    const int p0 = pixBase + tn * 16 + ln;
    const int p1 = p0 + 16;
    v8f ar0 = {}, ai0 = {}, ar1 = {}, ai1 = {};
    for (int kk = 0; kk < D_MODEL; kk += 4) {
      const int c0 = kk + hi * 2;
      const int c1 = c0 + 1;
      v2f fAr, fAi, fAiN;
      fAr.x = Wre[d * D_MODEL + c0];     // W is tiny: L2/WGP$ resident
      fAr.y = Wre[d * D_MODEL + c1];
      fAi.x = Wim[d * D_MODEL + c0];
      fAi.y = Wim[d * D_MODEL + c1];
      fAiN = -fAi;
      const size_t o0 = ((size_t)(b * D_MODEL + c0)) * IMG;
      const size_t o1 = ((size_t)(b * D_MODEL + c1)) * IMG;
      v2f fBr0, fBi0, fBr1, fBi1;
      fBr0.x = y2re[o0 + p0];            // coalesced across lanes (pix)
      fBr0.y = y2re[o1 + p0];
      fBi0.x = y2im[o0 + p0];
      fBi0.y = y2im[o1 + p0];
      fBr1.x = y2re[o0 + p1];
      fBr1.y = y2re[o1 + p1];
      fBi1.x = y2im[o0 + p1];
      fBi1.y = y2im[o1 + p1];
      ar0 = wmma_p(fAr, fBr0, ar0);
      ar0 = wmma_p(fAiN, fBi0, ar0);
      ai0 = wmma_p(fAr, fBi0, ai0);
      ai0 = wmma_p(fAi, fBr0, ai0);
      ar1 = wmma_p(fAr, fBr1, ar1);
      ar1 = wmma_p(fAiN, fBi1, ar1);
      ai1 = wmma_p(fAr, fBi1, ai1);
      ai1 = wmma_p(fAi, fBr1, ai1);
    }
    const int rbase = wave * 16 + hi * 8;
#pragma unroll
    for (int r = 0; r < 8; ++r) {
      const int drow = rbase + r;
      const size_t o = ((size_t)(b * D_MODEL + drow)) * IMG;
      out[2 * (o + p0)]     = ar0[r];   // complex64 interleaved
      out[2 * (o + p0) + 1] = ai0[r];
      out[2 * (o + p1)]     = ar1[r];
      out[2 * (o + p1) + 1] = ai1[r];
    }
  }
}

// ---------------------------------------------------------------------------
// Host launcher. Workspace layout (floats):
//   [0, 32768)  k_re (2,128,128) | [32768, 65536) k_im
//   then y1_re, y1_im, y2_re, y2_im  (each 4*128*128*128 floats). ~134.5 MB.
// ---------------------------------------------------------------------------
extern "C" void kernel_launch(void* const* d_in, const int* in_sizes, int n_in,
                              void* d_out, int out_size, void* d_ws,
                              size_t ws_size, hipStream_t stream) {
  const float* u_re   = (const float*)d_in[0];
  const float* u_im   = (const float*)d_in[1];
  const float* logA   = (const float*)d_in[2];
  const float* Aimag  = (const float*)d_in[3];
  const float* Bp_re  = (const float*)d_in[4];
  const float* Bp_im  = (const float*)d_in[5];
  const float* Cp_re  = (const float*)d_in[6];
  const float* Cp_im  = (const float*)d_in[7];
  const float* logdt  = (const float*)d_in[8];
  const float* W_re   = (const float*)d_in[9];
  const float* W_im   = (const float*)d_in[10];
  float* out = (float*)d_out;

  const size_t KSZ  = 2 * (size_t)D_MODEL * LEN;              // 32768
  const size_t PSZ  = (size_t)BATCH * D_MODEL * IMG;          // 8388608
  float* ws   = (float*)d_ws;
  float* k_re = ws;                  // (2, D, L)
  float* k_im = ws + KSZ;
  float* y1re = ws + 2 * KSZ;
  float* y1im = y1re + PSZ;
  float* y2re = y1im + PSZ;
  float* y2im = y2re + PSZ;
  float* k0re = k_re;                float* k0im = k_im;
  float* k1re = k_re + (size_t)D_MODEL * LEN;
  float* k1im = k_im + (size_t)D_MODEL * LEN;

  // Stage 1: SSM kernels (2*128 blocks, 128 threads)
  ssm_k_kernel<<<2 * D_MODEL, LEN, 0, stream>>>(
      logA, Aimag, logdt, Bp_re, Bp_im, Cp_re, Cp_im, k_re, k_im);

  // Stage 2: conv along L1 (one workgroup per (b,c) image)
  conv_rows_kernel<<<BATCH * D_MODEL, 256, 0, stream>>>(
      u_re, u_im, k0re, k0im, y1re, y1im);

  // Stage 3: conv along L2
  conv_cols_kernel<<<BATCH * D_MODEL, 256, 0, stream>>>(
      y1re, y1im, k1re, k1im, y2re, y2im);

  // Stage 4: channel mixing -> interleaved complex output
  mix_kernel<<<BATCH * D_MODEL, 256, 0, stream>>>(
      y2re, y2im, W_re, W_im, out);
}